// TransE_76768245448766
// MI455X (gfx1250) — compile-verified
//
#include <hip/hip_runtime.h>

#define N_ENT   40943
#define N_DIM   128
#define BATCH   32
#define LSTR    132                     // padded LDS row stride (floats) -> conflict-free WMMA reads
#define NTILES  ((N_ENT + 15) / 16)     // 2559 column tiles
#define WPB     4                       // waves per block

typedef __attribute__((ext_vector_type(2))) float v2f;
typedef __attribute__((ext_vector_type(8))) float v8f;

// ---------------------------------------------------------------------------
// Kernel 1: q[b] = ent_w[heads[b]] + rel_w[relations[b]];  qn[b] = ||q[b]||^2
// One wave per batch row (32 waves, 1 block).
// ---------------------------------------------------------------------------
__global__ __launch_bounds__(1024) void transe_prep(const int* __restrict__ heads,
                                                    const int* __restrict__ rels,
                                                    const float* __restrict__ ent_w,
                                                    const float* __restrict__ rel_w,
                                                    float* __restrict__ q,
                                                    float* __restrict__ qn) {
    int b    = threadIdx.x >> 5;   // 0..31
    int lane = threadIdx.x & 31;   // 0..31, each lane owns one float4 of the row
    int h = heads[b];
    int r = rels[b];
    float4 hv = ((const float4*)(ent_w + (size_t)h * N_DIM))[lane];
    float4 rv = ((const float4*)(rel_w + (size_t)r * N_DIM))[lane];
    float4 s;
    s.x = hv.x + rv.x; s.y = hv.y + rv.y; s.z = hv.z + rv.z; s.w = hv.w + rv.w;
    ((float4*)(q + b * N_DIM))[lane] = s;
    float p = s.x*s.x + s.y*s.y + s.z*s.z + s.w*s.w;
    #pragma unroll
    for (int off = 16; off > 0; off >>= 1) p += __shfl_xor(p, off, 32);
    if (lane == 0) qn[b] = p;
}

// ---------------------------------------------------------------------------
// Kernel 2: score[m, n] = qn[m] + ||ent_w[n]||^2 - 2 * (Q . E^T)[m, n]
// One wave per 16-entity column tile; each wave does both 16-row M tiles
// (B tile loaded from HBM exactly once, reused for both WMMA accumulators).
// ---------------------------------------------------------------------------
__global__ __launch_bounds__(128) void transe_score(const float* __restrict__ ent_w,
                                                    const float* __restrict__ q,
                                                    const float* __restrict__ qn,
                                                    float* __restrict__ out) {
    __shared__ float qs[BATCH * LSTR];        // 32 x 128 query matrix, padded
    __shared__ float bs[WPB][16 * LSTR];      // per-wave 16 x 128 entity tile

    int tid  = threadIdx.x;
    int wave = tid >> 5;
    int lane = tid & 31;

    // Stage Q (32x128) into LDS with padded rows (cooperative, coalesced).
    for (int i = tid; i < BATCH * (N_DIM / 4); i += 128) {
        int row = i >> 5;       // N_DIM/4 == 32 float4 chunks per row
        int c4  = i & 31;
        float4 v = ((const float4*)(q + row * N_DIM))[c4];
        float* dst = &qs[row * LSTR + c4 * 4];
        dst[0] = v.x; dst[1] = v.y; dst[2] = v.z; dst[3] = v.w;
    }
    __syncthreads();

    int tile = blockIdx.x * WPB + wave;
    if (tile >= NTILES) return;               // wave-uniform; EXEC stays full below
    int n0 = tile * 16;

    // Stage this wave's 16x128 entity tile (row-clamped on the last tile).
    float* bw = bs[wave];
    #pragma unroll 4
    for (int r = 0; r < 16; ++r) {
        int row = n0 + r;
        if (row > N_ENT - 1) row = N_ENT - 1;
        float4 v = ((const float4*)(ent_w + (size_t)row * N_DIM))[lane];
        float* dst = &bw[r * LSTR + lane * 4];
        dst[0] = v.x; dst[1] = v.y; dst[2] = v.z; dst[3] = v.w;
    }
    // Wave-local LDS producer->consumer: DS ops are in-order per wave and the
    // compiler inserts s_wait_dscnt for the dependent reads below.

    int lr = lane & 15;                       // column within tile / row M within half
    int lh = lane >> 4;                       // which K-pair half (WMMA f32 A/B stripe)

    // V_WMMA_F32_16X16X4_F32 over K = 128 in 32 steps; two M tiles share B.
    v8f acc0 = {};
    v8f acc1 = {};
    #pragma unroll
    for (int k = 0; k < N_DIM; k += 4) {
        v2f b  = *(const v2f*)&bw[lr * LSTR        + 2 * lh + k];  // B = E^T tile
        v2f a0 = *(const v2f*)&qs[lr * LSTR        + 2 * lh + k];  // Q rows 0..15
        v2f a1 = *(const v2f*)&qs[(16 + lr) * LSTR + 2 * lh + k];  // Q rows 16..31
        acc0 = __builtin_amdgcn_wmma_f32_16x16x4_f32(
                   false, a0, false, b, (short)0, acc0, false, false);
        acc1 = __builtin_amdgcn_wmma_f32_16x16x4_f32(
                   false, a1, false, b, (short)0, acc1, false, false);
    }

    // ||e||^2 for this lane's entity column: half-row per lane + cross-half add.
    float en = 0.0f;
    #pragma unroll
    for (int d = 0; d < 64; d += 2) {
        v2f t = *(const v2f*)&bw[lr * LSTR + lh * 64 + d];
        en += t.x * t.x + t.y * t.y;
    }
    en += __shfl_xor(en, 16, 32);

    // C layout: VGPR i -> M = i + 8*lh (+16 for second tile), N = n0 + lr.
    int n = n0 + lr;
    bool nok = (n < N_ENT);
    #pragma unroll
    for (int i = 0; i < 8; ++i) {
        int m0 = i + 8 * lh;
        int m1 = 16 + m0;
        if (nok) {
            out[(size_t)m0 * N_ENT + n] = qn[m0] + en - 2.0f * acc0[i];
            out[(size_t)m1 * N_ENT + n] = qn[m1] + en - 2.0f * acc1[i];
        }
    }
}

// ---------------------------------------------------------------------------
extern "C" void kernel_launch(void* const* d_in, const int* in_sizes, int n_in,
                              void* d_out, int out_size, void* d_ws, size_t ws_size,
                              hipStream_t stream) {
    (void)in_sizes; (void)n_in; (void)out_size; (void)ws_size;
    const int*   heads = (const int*)d_in[0];    // [32,1] int
    const int*   rels  = (const int*)d_in[1];    // [32]   int
    const float* ent_w = (const float*)d_in[2];  // [40943,128] f32
    const float* rel_w = (const float*)d_in[3];  // [18,128]    f32
    float* out = (float*)d_out;                  // [32,40943]  f32

    float* q  = (float*)d_ws;                    // 32*128 floats
    float* qn = q + BATCH * N_DIM;               // 32 floats

    transe_prep<<<1, 1024, 0, stream>>>(heads, rels, ent_w, rel_w, q, qn);

    int nblocks = (NTILES + WPB - 1) / WPB;      // 640
    transe_score<<<nblocks, 128, 0, stream>>>(ent_w, q, qn, out);
}